// LKalmanFilter_36773509989079
// MI455X (gfx1250) — compile-verified
//
#include <hip/hip_runtime.h>

// ---------------------------------------------------------------------------
// Layerwise Kalman update, MI455X (gfx1250).
//   pass1: PH = P@H and HtP = H^T@P via V_WMMA_F32_16X16X4_F32, async LDS staging
//   pass2: A = 1/(B*LAM + dot), K = A*PH, new_w = w + K*error
//   pass3: P_new = 0.5*(P + P^T - K HtP^T - HtP K^T)/LAM, tile-pair symmetric
// ---------------------------------------------------------------------------

#define LAMK 0.98f
#define BQ   32
#define NQ   2048

typedef __attribute__((ext_vector_type(2))) float v2f;
typedef __attribute__((ext_vector_type(8))) float v8f;
typedef __attribute__((ext_vector_type(4))) int   v4i;

#if defined(__has_builtin)
#  if __has_builtin(__builtin_amdgcn_global_load_async_to_lds_b128)
#    define HAVE_ASYNC_LDS 1
#  endif
#  if __has_builtin(__builtin_amdgcn_s_wait_asynccnt)
#    define HAVE_WAIT_ASYNC 1
#  endif
#endif

typedef __attribute__((address_space(1))) v4i* g_v4i_p;
typedef __attribute__((address_space(3))) v4i* l_v4i_p;

__device__ __forceinline__ void async_copy16(const float* src, float* dst) {
#if defined(HAVE_ASYNC_LDS)
    __builtin_amdgcn_global_load_async_to_lds_b128(
        (g_v4i_p)(src), (l_v4i_p)(dst), 0, 0);
#else
    *(float4*)dst = *(const float4*)src;
#endif
}

__device__ __forceinline__ void async_wait_all() {
#if defined(HAVE_WAIT_ASYNC)
    __builtin_amdgcn_s_wait_asynccnt(0);
#elif defined(HAVE_ASYNC_LDS)
    asm volatile("s_wait_asynccnt 0" ::: "memory");
#endif
}

// ---------------------------------------------------------------------------
// init: zero the atomic accumulators (re-run every launch -> graph safe)
// ---------------------------------------------------------------------------
__global__ void kf_init(float* dotAcc, float* HtP) {
    int i = blockIdx.x * 256 + threadIdx.x;
    if (i == 0) dotAcc[0] = 0.0f;
    if (i < BQ * NQ) HtP[i] = 0.0f;
}

// ---------------------------------------------------------------------------
// pass1: one workgroup (8 waves) per 128-row panel of one block.
// Each wave owns 16 rows and runs dual WMMA (PH + HtP) over 128-col tiles
// staged in LDS (double-buffered async copies).
// LDS floats: 2*128*132 (stage) + 2048 (H) + 2048 (HtP acc) = 37888
// ---------------------------------------------------------------------------
#define P1_PITCH 132
#define P1_LDSF  (2 * 128 * P1_PITCH + NQ + NQ)

__device__ __forceinline__ void stage_tile128(const float* Pb, int rowBase, int s,
                                              float* buf, int t) {
#pragma unroll 4
    for (int k = 0; k < 16; ++k) {
        int f   = t + k * 256;      // 4096 float4 per tile
        int row = f >> 5;           // 32 float4 per row
        int c4  = (f & 31) << 2;    // col in floats
        const float* src = Pb + (size_t)(rowBase + row) * NQ + s * 128 + c4;
        async_copy16(src, buf + row * P1_PITCH + c4);
    }
}

__global__ void __launch_bounds__(256) kf_pass1(const float* __restrict__ P,
                                                const float* __restrict__ H,
                                                float* __restrict__ PH,
                                                float* __restrict__ HtP,
                                                float* __restrict__ dotAcc) {
    extern __shared__ float lds[];
    float* stage0 = lds;
    float* stage1 = lds + 128 * P1_PITCH;
    float* hH     = lds + 2 * 128 * P1_PITCH;   // H[b,:] (serves rows and cols)
    float* htp    = hH + NQ;                    // workgroup HtP accumulator

    const int wg    = blockIdx.x;
    const int b     = wg >> 4;      // 16 panels per block
    const int panel = wg & 15;
    const int t     = threadIdx.x;
    const int lane  = t & 31;
    const int wave  = t >> 5;
    const int sel   = (lane >> 4) & 1;  // lane-half: selects K {0,1} vs {2,3}
    const int laneM = lane & 15;

    const float* Pb = P + (size_t)b * NQ * NQ;
    const int rowBase = panel * 128;

    for (int j = t; j < NQ; j += 256) {
        htp[j] = 0.0f;
        hH[j]  = H[b * NQ + j];
    }
    __syncthreads();

    // dir2 A operand: row-H chunk broadcast across all M rows.
    const int i0 = rowBase + wave * 16;         // this wave's first row (in block)
    v2f a2[4];
#pragma unroll
    for (int c = 0; c < 4; ++c) {
        int idx = i0 + 4 * c + 2 * sel;
        v2f tmp = { hH[idx], hH[idx + 1] };
        a2[c] = tmp;
    }

    v8f d1 = {0.f, 0.f, 0.f, 0.f, 0.f, 0.f, 0.f, 0.f};   // PH accumulator (col 0 of D)
    const v2f vzero = {0.f, 0.f};

    stage_tile128(Pb, rowBase, 0, stage0, t);

    for (int s = 0; s < 16; ++s) {
        async_wait_all();
        __syncthreads();                         // stage s visible to all waves
        float* buf  = (s & 1) ? stage1 : stage0;
        float* nbuf = (s & 1) ? stage0 : stage1;
        if (s + 1 < 16) stage_tile128(Pb, rowBase, s + 1, nbuf, t);

        const int colBase = s * 128;
        const float* rowp = buf + (wave * 16 + laneM) * P1_PITCH;

        for (int jt = 0; jt < 8; ++jt) {
            v8f d2 = {0.f, 0.f, 0.f, 0.f, 0.f, 0.f, 0.f, 0.f};
#pragma unroll
            for (int c = 0; c < 4; ++c) {
                const int cl = jt * 16 + 4 * c + 2 * sel;
                // ---- dir1: D(16x16) += Ptile(16x4-chunk) x [h | 0 ... 0]
                v2f a1 = { rowp[cl], rowp[cl + 1] };
                v2f hb = { hH[colBase + cl], hH[colBase + cl + 1] };
                v2f b1 = (laneM == 0) ? hb : vzero;
                d1 = __builtin_amdgcn_wmma_f32_16x16x4_f32(
                        false, a1, false, b1, (short)0, d1, false, false);
                // ---- dir2: D(16x16) += broadcast(hrow-chunk) x Ptile-rows
                const float* r0 = buf + (wave * 16 + 4 * c + 2 * sel) * P1_PITCH
                                      + jt * 16 + laneM;
                v2f b2 = { r0[0], r0[P1_PITCH] };
                d2 = __builtin_amdgcn_wmma_f32_16x16x4_f32(
                        false, a2[c], false, b2, (short)0, d2, false, false);
            }
            // HtP partial for 16 cols lives in D row 0: VGPR0, lanes 0..15
            if (lane < 16)
                atomicAdd(&htp[colBase + jt * 16 + laneM], d2[0]);
        }
        __syncthreads();                         // all waves done with buf
    }

    // PH = D column 0: lane 0 holds m=0..7 (v0..7), lane 16 holds m=8..15
    if (laneM == 0) {
        const int mBase = i0 + 8 * sel;
        float dp = 0.0f;
#pragma unroll
        for (int v = 0; v < 8; ++v) {
            float ph = d1[v];
            PH[b * NQ + mBase + v] = ph;
            dp += ph * hH[mBase + v];
        }
        atomicAdd(dotAcc, dp);                   // sum_i H_i * PH_i
    }

    __syncthreads();
    for (int j = t; j < NQ; j += 256)
        atomicAdd(&HtP[b * NQ + j], htp[j]);
}

// ---------------------------------------------------------------------------
// pass2: K = PH / (B*LAM + dot);  new_w = w + K*error
// ---------------------------------------------------------------------------
__global__ void kf_pass2(const float* __restrict__ PH, const float* __restrict__ w,
                         const float* __restrict__ err, const float* __restrict__ dotAcc,
                         float* __restrict__ K, float* __restrict__ outW) {
    int i = blockIdx.x * 256 + threadIdx.x;
    if (i >= BQ * NQ) return;
    float A = 1.0f / ((float)BQ * LAMK + dotAcc[0]);
    float k = PH[i] * A;
    K[i]    = k;
    outW[i] = w[i] + k * err[0];
}

// ---------------------------------------------------------------------------
// pass3: symmetric rank-1 update, tile-pair (I,J)/(J,I) of 64x64 so every
// P element is read once; transpose through padded LDS.
// LDS floats: 2*64*76 + 4*64 = 9984
// ---------------------------------------------------------------------------
#define P3_PITCH 76
#define P3_LDSF  (2 * 64 * P3_PITCH + 4 * 64)

__global__ void __launch_bounds__(256) kf_pass3(const float* __restrict__ P,
                                                const float* __restrict__ K,
                                                const float* __restrict__ HtP,
                                                float* __restrict__ Pout) {
    extern __shared__ float lds[];
    float* ldsT = lds;                       // P[J-tile][I-tile], row-major
    float* ldsS = lds + 64 * P3_PITCH;       // results for transposed writeback
    float* kI   = ldsS + 64 * P3_PITCH;
    float* kJ   = kI + 64;
    float* hI   = kJ + 64;
    float* hJ   = hI + 64;

    int wg = blockIdx.x;
    int b  = wg / 528;                       // 528 = 32*33/2 tile pairs
    int p  = wg % 528;
    int I  = 0;
    while (p >= (32 - I)) { p -= (32 - I); ++I; }
    int J = I + p;                           // I <= J

    const int t = threadIdx.x;
    const size_t base = (size_t)b * NQ * NQ;
    const float c1 = 0.5f / LAMK;

    if (t < 64) {
        kI[t] = K[b * NQ + I * 64 + t];
        hI[t] = HtP[b * NQ + I * 64 + t];
        kJ[t] = K[b * NQ + J * 64 + t];
        hJ[t] = HtP[b * NQ + J * 64 + t];
    }
    // stage tile (J,I): rows J*64.., cols I*64..
#pragma unroll
    for (int k = 0; k < 4; ++k) {
        int f  = t + k * 256;                // 1024 float4
        int r  = f >> 4;                     // 16 float4 per row
        int c4 = (f & 15) << 2;
        const float* src = P + base + (size_t)(J * 64 + r) * NQ + I * 64 + c4;
        async_copy16(src, ldsT + r * P3_PITCH + c4);
    }
    async_wait_all();
    __syncthreads();

#pragma unroll
    for (int g = 0; g < 4; ++g) {
        int r  = (t >> 4) + g * 16;          // row in (I,J) tile
        int c0 = (t & 15) << 2;
        float4 pd = *(const float4*)(P + base + (size_t)(I * 64 + r) * NQ + J * 64 + c0);
        float ki = kI[r], hi = hI[r];
        float o[4];
#pragma unroll
        for (int q = 0; q < 4; ++q) {
            int c = c0 + q;
            float pdq = (q == 0) ? pd.x : (q == 1) ? pd.y : (q == 2) ? pd.z : pd.w;
            float pt  = ldsT[c * P3_PITCH + r];          // P[j][i]
            float s   = c1 * ((pdq + pt) - (ki * hJ[c] + kJ[c] * hi));
            o[q] = s;
            ldsS[r * P3_PITCH + c] = s;
        }
        float4 ov = { o[0], o[1], o[2], o[3] };
        *(float4*)(Pout + base + (size_t)(I * 64 + r) * NQ + J * 64 + c0) = ov;
    }

    if (I != J) {                            // uniform branch across workgroup
        __syncthreads();
#pragma unroll
        for (int g = 0; g < 4; ++g) {
            int r  = (t >> 4) + g * 16;      // row in (J,I) tile
            int c0 = (t & 15) << 2;
            float4 ov = { ldsS[(c0 + 0) * P3_PITCH + r],
                          ldsS[(c0 + 1) * P3_PITCH + r],
                          ldsS[(c0 + 2) * P3_PITCH + r],
                          ldsS[(c0 + 3) * P3_PITCH + r] };
            *(float4*)(Pout + base + (size_t)(J * 64 + r) * NQ + I * 64 + c0) = ov;
        }
    }
}

// ---------------------------------------------------------------------------
extern "C" void kernel_launch(void* const* d_in, const int* in_sizes, int n_in,
                              void* d_out, int out_size, void* d_ws, size_t ws_size,
                              hipStream_t stream) {
    (void)in_sizes; (void)n_in; (void)out_size; (void)ws_size;
    const float* P   = (const float*)d_in[0];
    const float* H   = (const float*)d_in[1];
    const float* w   = (const float*)d_in[2];
    const float* err = (const float*)d_in[3];

    float* outW = (float*)d_out;             // [B*N]
    float* Pout = outW + BQ * NQ;            // [B*N*N]

    float* wsf    = (float*)d_ws;
    float* dotAcc = wsf;                     // 1 (padded to 64)
    float* PHbuf  = wsf + 64;                // B*N
    float* HtPbuf = PHbuf + BQ * NQ;         // B*N
    float* Kbuf   = HtPbuf + BQ * NQ;        // B*N

    const int nv = BQ * NQ;
    kf_init<<<(nv + 255) / 256, 256, 0, stream>>>(dotAcc, HtPbuf);
    kf_pass1<<<dim3(BQ * 16), dim3(256), P1_LDSF * sizeof(float), stream>>>(
        P, H, PHbuf, HtPbuf, dotAcc);
    kf_pass2<<<(nv + 255) / 256, 256, 0, stream>>>(PHbuf, w, err, dotAcc, Kbuf, outW);
    kf_pass3<<<dim3(BQ * 528), dim3(256), P3_LDSF * sizeof(float), stream>>>(
        P, Kbuf, HtPbuf, Pout);
}